// LinearAttention_9199819948514
// MI455X (gfx1250) — compile-verified
//
#include <hip/hip_runtime.h>
#include <math.h>

// Linear attention (elu+1 feature map) for n=8, s=8192, h=8, d=m=64, fp32.
// Strategy: two GEMM phases on V_WMMA_F32_16X16X4_F32 (native fp32 WMMA,
// exact fp32 accumulate). Memory-bound problem (~537MB @ 23.3TB/s ~ 23us);
// fp32 WMMA keeps full reference precision.

typedef __attribute__((ext_vector_type(2))) float v2f;
typedef __attribute__((ext_vector_type(8))) float v8f;

#define NB   8
#define SEQ  8192
#define NHD  8
#define DD   64
#define RS   (NHD * DD)          // 512 floats: stride between s-rows for fixed (n,h)
#define LA_EPS 1e-6f

// workspace layout (floats): KV[64 nh][64 m][64 d], then Ksum[64 nh][64 d]
#define KV_FLOATS   (64 * 64 * 64)
#define KSUM_OFF    KV_FLOATS
#define WS_FLOATS   (KV_FLOATS + 64 * 64)

__device__ __forceinline__ float fmap(float x) {
    // elu(x) + 1 = x>0 ? x+1 : exp(x)
    return x > 0.0f ? x + 1.0f : __expf(x);
}

__device__ __forceinline__ v8f wmma_f32(v2f a, v2f b, v8f c) {
    // D = A(16x4) x B(4x16) + C(16x16), all fp32
    return __builtin_amdgcn_wmma_f32_16x16x4_f32(false, a, false, b,
                                                 (short)0, c, false, false);
}

__global__ void la_init(float* __restrict__ ws, int n) {
    int i = blockIdx.x * 256 + threadIdx.x;
    if (i < n) ws[i] = 0.0f;
}

// ---------------------------------------------------------------------------
// Phase 1: KV[m,d] = sum_s V[s,m] * fmap(K[s,d]);  Ksum[d] = sum_s fmap(K[s,d])
// grid (64 nh, 16 s-chunks), 256 threads (8 waves).
// waves: mt = wave&3 owns m-rows [16mt,16mt+16); kg = wave>>2 splits s-chunk.
// ---------------------------------------------------------------------------
__global__ __launch_bounds__(256) void la_phase1(const float* __restrict__ K,
                                                 const float* __restrict__ V,
                                                 float* __restrict__ ws) {
    __shared__ float skv[64 * 64];   // KV tile accumulator [m][d]

    const int tid   = threadIdx.x;
    const int lane  = tid & 31;
    const int wave  = tid >> 5;
    const int lm    = lane & 15;
    const int lhalf = lane >> 4;     // 0: K-rows {0,1}; 1: K-rows {2,3}
    const int mt    = wave & 3;
    const int kg    = wave >> 2;

    const int nh = blockIdx.x;
    const int nn = nh >> 3;
    const int hh = nh & 7;
    const int s0 = blockIdx.y * 512 + kg * 256;

#pragma unroll
    for (int i = 0; i < 16; ++i) skv[tid + 256 * i] = 0.0f;
    __syncthreads();

    const size_t base = (((size_t)nn * SEQ + s0) * NHD + hh) * DD;
    const float* Kp = K + base;
    const float* Vp = V + base;

    const int m0    = 16 * mt;
    const int roff0 = (2 * lhalf) * RS;   // V/K row for A/B VGPR0 (K=2*lhalf)
    const int roff1 = roff0 + RS;         // row for VGPR1 (K=2*lhalf+1)

    v8f acc0 = {}, acc1 = {}, acc2 = {}, acc3 = {};
    float ks0 = 0.f, ks1 = 0.f, ks2 = 0.f, ks3 = 0.f;

    for (int it = 0; it < 64; ++it) {    // 64 iters x 4 s-rows = 256 rows/wave
        const float* Kb = Kp + (size_t)it * (4 * RS);
        const float* Vb = Vp + (size_t)it * (4 * RS);

        // A = V chunk: A[m,k] = V[s0+k, m]  (ISA 32-bit 16x4 A layout)
        v2f a;
        a.x = Vb[roff0 + m0 + lm];
        a.y = Vb[roff1 + m0 + lm];

        // B = fmap(K) chunk: B[k,d] = Kf[s0+k, d]; rows striped across lanes,
        // VGPR0 = rows {0,2}, VGPR1 = rows {1,3} split by lane half.
        v2f b0, b1, b2, b3;
        b0.x = fmap(Kb[roff0 +  0 + lm]);  b0.y = fmap(Kb[roff1 +  0 + lm]);
        b1.x = fmap(Kb[roff0 + 16 + lm]);  b1.y = fmap(Kb[roff1 + 16 + lm]);
        b2.x = fmap(Kb[roff0 + 32 + lm]);  b2.y = fmap(Kb[roff1 + 32 + lm]);
        b3.x = fmap(Kb[roff0 + 48 + lm]);  b3.y = fmap(Kb[roff1 + 48 + lm]);

        if (mt == 0) {   // Ksum piggy-back (only one m-strip wave per kg)
            ks0 += b0.x + b0.y;  ks1 += b1.x + b1.y;
            ks2 += b2.x + b2.y;  ks3 += b3.x + b3.y;
        }

        acc0 = wmma_f32(a, b0, acc0);
        acc1 = wmma_f32(a, b1, acc1);
        acc2 = wmma_f32(a, b2, acc2);
        acc3 = wmma_f32(a, b3, acc3);
    }

    // deposit partial C tiles into LDS (ds_add_f32 handles kg collisions)
#pragma unroll
    for (int j = 0; j < 8; ++j) {
        const int row = m0 + j + 8 * lhalf;  // C layout: VGPR j -> M = j | j+8
        atomicAdd(&skv[row * 64 +  0 + lm], acc0[j]);
        atomicAdd(&skv[row * 64 + 16 + lm], acc1[j]);
        atomicAdd(&skv[row * 64 + 32 + lm], acc2[j]);
        atomicAdd(&skv[row * 64 + 48 + lm], acc3[j]);
    }

    if (mt == 0) {
        float* ksw = ws + KSUM_OFF + nh * 64;
        atomicAdd(&ksw[ 0 + lm], ks0);
        atomicAdd(&ksw[16 + lm], ks1);
        atomicAdd(&ksw[32 + lm], ks2);
        atomicAdd(&ksw[48 + lm], ks3);
    }
    __syncthreads();

    float* kvw = ws + (size_t)nh * 4096;
#pragma unroll
    for (int i = 0; i < 16; ++i)
        atomicAdd(&kvw[tid + 256 * i], skv[tid + 256 * i]);
}

// ---------------------------------------------------------------------------
// Phase 2: out[l,m] = (1/(Qf[l,:].Ksum + eps)) * sum_d Qf[l,d] * KV[m,d]
// grid (64 nh, 64 l-blocks of 128 rows), 256 threads; each wave does 16 rows.
// ---------------------------------------------------------------------------
__global__ __launch_bounds__(256) void la_phase2(const float* __restrict__ Q,
                                                 const float* __restrict__ ws,
                                                 float* __restrict__ out) {
    __shared__ float skv[64 * 64];   // KV[m][d]
    __shared__ float sks[64];        // Ksum[d]

    const int tid   = threadIdx.x;
    const int lane  = tid & 31;
    const int wave  = tid >> 5;
    const int lm    = lane & 15;
    const int lhalf = lane >> 4;

    const int nh = blockIdx.x;
    const int nn = nh >> 3;
    const int hh = nh & 7;

    const float* kvw = ws + (size_t)nh * 4096;
#pragma unroll
    for (int i = 0; i < 16; ++i) skv[tid + 256 * i] = kvw[tid + 256 * i];
    if (tid < 64) sks[tid] = ws[KSUM_OFF + nh * 64 + tid];
    __syncthreads();

    const int l0 = (blockIdx.y * 8 + wave) * 16;
    const float* Qrow = Q + (((size_t)nn * SEQ + l0 + lm) * NHD + hh) * DD;
    const int k2 = 2 * lhalf;

    v8f acc0 = {}, acc1 = {}, acc2 = {}, acc3 = {};
    float denom = 0.0f;

    for (int kk = 0; kk < 64; kk += 4) {
        // A[l,k] = Qf[l0+l, kk+k]
        v2f a;
        a.x = fmap(Qrow[kk + k2]);
        a.y = fmap(Qrow[kk + k2 + 1]);

        // normalizer partial: this lane covers d = kk+k2, kk+k2+1
        denom += a.x * sks[kk + k2] + a.y * sks[kk + k2 + 1];

        // B[k,n] = KV[n, kk+k]  (KV^T tile columns from LDS)
        v2f b0, b1, b2, b3;
        b0.x = skv[( 0 + lm) * 64 + kk + k2];  b0.y = skv[( 0 + lm) * 64 + kk + k2 + 1];
        b1.x = skv[(16 + lm) * 64 + kk + k2];  b1.y = skv[(16 + lm) * 64 + kk + k2 + 1];
        b2.x = skv[(32 + lm) * 64 + kk + k2];  b2.y = skv[(32 + lm) * 64 + kk + k2 + 1];
        b3.x = skv[(48 + lm) * 64 + kk + k2];  b3.y = skv[(48 + lm) * 64 + kk + k2 + 1];

        acc0 = wmma_f32(a, b0, acc0);
        acc1 = wmma_f32(a, b1, acc1);
        acc2 = wmma_f32(a, b2, acc2);
        acc3 = wmma_f32(a, b3, acc3);
    }

    // lane halves each saw half the d values -> combine across lane^16
    denom += __shfl_xor(denom, 16, 32);
    const float rcp = 1.0f / (denom + LA_EPS);

    float* Obase = out + (((size_t)nn * SEQ + l0) * NHD + hh) * DD;
#pragma unroll
    for (int j = 0; j < 8; ++j) {
        const int row = j + 8 * lhalf;                 // C layout row
        const float zr = __shfl(rcp, row, 32);         // rcp lives in lane lm==row
        float* Orow = Obase + (size_t)row * RS;
        Orow[ 0 + lm] = acc0[j] * zr;
        Orow[16 + lm] = acc1[j] * zr;
        Orow[32 + lm] = acc2[j] * zr;
        Orow[48 + lm] = acc3[j] * zr;
    }
}

extern "C" void kernel_launch(void* const* d_in, const int* in_sizes, int n_in,
                              void* d_out, int out_size, void* d_ws, size_t ws_size,
                              hipStream_t stream) {
    const float* Q = (const float*)d_in[0];
    const float* K = (const float*)d_in[1];
    const float* V = (const float*)d_in[2];
    // d_in[3] (h=128) and d_in[4] (w=64) are unused by the reference.
    float* out = (float*)d_out;
    float* ws  = (float*)d_ws;   // needs WS_FLOATS*4 ~ 1.04 MB

    la_init<<<dim3((WS_FLOATS + 255) / 256), dim3(256), 0, stream>>>(ws, WS_FLOATS);
    la_phase1<<<dim3(64, SEQ / 512), dim3(256), 0, stream>>>(K, V, ws);
    la_phase2<<<dim3(64, SEQ / 128), dim3(256), 0, stream>>>(Q, ws, out);
}